// ImageNormalization2D_9156870275416
// MI455X (gfx1250) — compile-verified
//
#include <hip/hip_runtime.h>
#include <stdint.h>

// Problem constants (reference: x = (16, 1024, 1024, 2) fp32 NHWC, 61x61 box)
#define N_   16
#define H_   1024
#define W_   1024
#define FS   61
#define RAD  30
#define INV_DENOM (1.0f / 7442.0f)   // 1/(61*61*2)
#define EPS_ 1e-7f

// ---- tiny float2 helper (avoid dependence on HIP vector operators) ----
struct f2 { float x, y; };
__device__ __forceinline__ f2 f2add(f2 a, f2 b) { return {a.x + b.x, a.y + b.y}; }
__device__ __forceinline__ f2 f2sub(f2 a, f2 b) { return {a.x - b.x, a.y - b.y}; }
__device__ __forceinline__ f2 f2sq (f2 a)       { return {a.x * a.x, a.y * a.y}; }

// ---- CDNA5 async global->LDS staging path (guarded) ----
#if defined(__has_builtin)
#  if __has_builtin(__builtin_amdgcn_global_load_async_to_lds_b128) && \
      __has_builtin(__builtin_amdgcn_s_wait_asynccnt)
#    define USE_ASYNC_LDS 1
#  endif
#endif
#ifndef USE_ASYNC_LDS
#  define USE_ASYNC_LDS 0
#endif

#if USE_ASYNC_LDS
typedef int v4i __attribute__((ext_vector_type(4)));
typedef __attribute__((address_space(1))) v4i g_v4i;   // global int4
typedef __attribute__((address_space(3))) v4i s_v4i;   // LDS int4
#endif

// Copy one image row (1024 f2 = 8KB = 512 x b128) from global into LDS.
// 'nthreads' must divide 512 with a uniform trip count per wave.
__device__ __forceinline__ void stage_row(const float* __restrict__ gsrc,
                                          f2* lds_dst, int tid, int nthreads) {
#if USE_ASYNC_LDS
  float* ldsf = (float*)lds_dst;
  for (int i = tid; i < 512; i += nthreads) {
    __builtin_amdgcn_global_load_async_to_lds_b128(
        (g_v4i*)(gsrc + i * 4), (s_v4i*)(ldsf + i * 4), 0, 0);
  }
  __builtin_amdgcn_s_wait_asynccnt(0);
#else
  const float4* s4 = (const float4*)gsrc;
  float4* d4 = (float4*)lds_dst;
  for (int i = tid; i < 512; i += nthreads) d4[i] = s4[i];
#endif
}

// ---------------------------------------------------------------------------
// Pass 1: horizontal 61-sum of x -> t1.  One block per (n,y) row, 64 threads,
// each thread produces 16 consecutive outputs with a sliding-window sum.
// ---------------------------------------------------------------------------
__global__ __launch_bounds__(64) void hbox1_kernel(const float* __restrict__ x,
                                                   float* __restrict__ t1) {
  __shared__ f2 srow[W_ + 2 * RAD];  // zero-padded input row
  __shared__ f2 orow[W_];            // staged output row (for coalesced store)
  const int row = blockIdx.x;        // n*H + y
  const int tid = threadIdx.x;
  const float* src = x + (size_t)row * (W_ * 2);

  for (int i = tid; i < RAD; i += 64) {        // zero both pads
    srow[i] = {0.f, 0.f};
    srow[RAD + W_ + i] = {0.f, 0.f};
  }
  stage_row(src, srow + RAD, tid, 64);         // srow+RAD is 240B offset (16B aligned)
  __syncthreads();

  const int x0 = tid * 16;
  f2 s = {0.f, 0.f};
  for (int k = 0; k < FS; ++k) s = f2add(s, srow[x0 + k]);   // window [x0-30, x0+30]
  orow[x0] = s;
  #pragma unroll
  for (int i = 1; i < 16; ++i) {
    s = f2add(s, srow[x0 + FS - 1 + i]);
    s = f2sub(s, srow[x0 + i - 1]);
    orow[x0 + i] = s;
  }
  __syncthreads();

  float4* dst4 = (float4*)(t1 + (size_t)row * (W_ * 2));
  const float4* s4 = (const float4*)orow;
  for (int i = tid; i < 512; i += 64) dst4[i] = s4[i];
}

// ---------------------------------------------------------------------------
// Pass 2: vertical 61-sum of t1, then out = x - sum/7442 -> outs (d_out stage).
// Column strips: grid (xblk=4, yseg=4, n=16), 256 threads = 256 f2 columns.
// ---------------------------------------------------------------------------
__global__ __launch_bounds__(256) void vbox1_kernel(const float* __restrict__ t1,
                                                    const float* __restrict__ x,
                                                    float* __restrict__ outs) {
  const int tid  = threadIdx.x;
  const int xcol = blockIdx.x * 256 + tid;
  const int y0   = blockIdx.y * 256;
  const size_t plane = (size_t)blockIdx.z * H_ * W_;   // in f2 units
  const f2* t   = (const f2*)t1   + plane + xcol;
  const f2* xin = (const f2*)x    + plane + xcol;
  f2*       o   = (f2*)outs       + plane + xcol;

  f2 s = {0.f, 0.f};
  const int ylo = (y0 - RAD < 0) ? 0 : y0 - RAD;
  const int yhi = (y0 + RAD > H_ - 1) ? H_ - 1 : y0 + RAD;
  for (int yy = ylo; yy <= yhi; ++yy) s = f2add(s, t[(size_t)yy * W_]);

  for (int j = 0; j < 256; ++j) {
    const int y = y0 + j;
    __builtin_prefetch(&t[(size_t)(y + 40 <= H_ - 1 ? y + 40 : H_ - 1) * W_], 0, 1);
    const f2 xv = xin[(size_t)y * W_];
    o[(size_t)y * W_] = { xv.x - s.x * INV_DENOM, xv.y - s.y * INV_DENOM };
    const int ya = y + RAD + 1, yr = y - RAD;
    if (ya <= H_ - 1) s = f2add(s, t[(size_t)ya * W_]);
    if (yr >= 0)      s = f2sub(s, t[(size_t)yr * W_]);
  }
}

// ---------------------------------------------------------------------------
// Pass 3: horizontal 61-sums of out and out^2 -> tA, tB (single sliding pass).
// ---------------------------------------------------------------------------
__global__ __launch_bounds__(64) void hbox2_kernel(const float* __restrict__ outs,
                                                   float* __restrict__ tA,
                                                   float* __restrict__ tB) {
  __shared__ f2 srow[W_ + 2 * RAD];
  __shared__ f2 oa[W_];
  __shared__ f2 ob[W_];
  const int row = blockIdx.x;
  const int tid = threadIdx.x;
  const float* src = outs + (size_t)row * (W_ * 2);

  for (int i = tid; i < RAD; i += 64) {
    srow[i] = {0.f, 0.f};
    srow[RAD + W_ + i] = {0.f, 0.f};
  }
  stage_row(src, srow + RAD, tid, 64);
  __syncthreads();

  const int x0 = tid * 16;
  f2 s1 = {0.f, 0.f}, s2 = {0.f, 0.f};
  for (int k = 0; k < FS; ++k) {
    const f2 v = srow[x0 + k];
    s1 = f2add(s1, v);
    s2 = f2add(s2, f2sq(v));
  }
  oa[x0] = s1; ob[x0] = s2;
  #pragma unroll
  for (int i = 1; i < 16; ++i) {
    const f2 vn = srow[x0 + FS - 1 + i];
    const f2 vo = srow[x0 + i - 1];
    s1 = f2sub(f2add(s1, vn), vo);
    s2 = f2sub(f2add(s2, f2sq(vn)), f2sq(vo));
    oa[x0 + i] = s1; ob[x0 + i] = s2;
  }
  __syncthreads();

  float4* dA = (float4*)(tA + (size_t)row * (W_ * 2));
  float4* dB = (float4*)(tB + (size_t)row * (W_ * 2));
  const float4* a4 = (const float4*)oa;
  const float4* b4 = (const float4*)ob;
  for (int i = tid; i < 512; i += 64) { dA[i] = a4[i]; dB[i] = b4[i]; }
}

// ---------------------------------------------------------------------------
// Pass 4: vertical 61-sums of tA/tB; std = sqrt(q - m^2) + eps; outs /= std
// (in place: each thread only touches its own element of outs).
// ---------------------------------------------------------------------------
__global__ __launch_bounds__(256) void vbox2_kernel(const float* __restrict__ tA,
                                                    const float* __restrict__ tB,
                                                    float* __restrict__ outs) {
  const int tid  = threadIdx.x;
  const int xcol = blockIdx.x * 256 + tid;
  const int y0   = blockIdx.y * 256;
  const size_t plane = (size_t)blockIdx.z * H_ * W_;
  const f2* ta = (const f2*)tA + plane + xcol;
  const f2* tb = (const f2*)tB + plane + xcol;
  f2*       o  = (f2*)outs     + plane + xcol;

  f2 sa = {0.f, 0.f}, sb = {0.f, 0.f};
  const int ylo = (y0 - RAD < 0) ? 0 : y0 - RAD;
  const int yhi = (y0 + RAD > H_ - 1) ? H_ - 1 : y0 + RAD;
  for (int yy = ylo; yy <= yhi; ++yy) {
    sa = f2add(sa, ta[(size_t)yy * W_]);
    sb = f2add(sb, tb[(size_t)yy * W_]);
  }

  for (int j = 0; j < 256; ++j) {
    const int y = y0 + j;
    __builtin_prefetch(&ta[(size_t)(y + 40 <= H_ - 1 ? y + 40 : H_ - 1) * W_], 0, 1);
    __builtin_prefetch(&tb[(size_t)(y + 40 <= H_ - 1 ? y + 40 : H_ - 1) * W_], 0, 1);
    const f2 m = { sa.x * INV_DENOM, sa.y * INV_DENOM };
    const f2 q = { sb.x * INV_DENOM, sb.y * INV_DENOM };
    const f2 sd = { sqrtf(q.x - m.x * m.x) + EPS_, sqrtf(q.y - m.y * m.y) + EPS_ };
    f2 ov = o[(size_t)y * W_];
    o[(size_t)y * W_] = { ov.x / sd.x, ov.y / sd.y };
    const int ya = y + RAD + 1, yr = y - RAD;
    if (ya <= H_ - 1) { sa = f2add(sa, ta[(size_t)ya * W_]); sb = f2add(sb, tb[(size_t)ya * W_]); }
    if (yr >= 0)      { sa = f2sub(sa, ta[(size_t)yr * W_]); sb = f2sub(sb, tb[(size_t)yr * W_]); }
  }
}

// ---------------------------------------------------------------------------
extern "C" void kernel_launch(void* const* d_in, const int* in_sizes, int n_in,
                              void* d_out, int out_size, void* d_ws, size_t ws_size,
                              hipStream_t stream) {
  (void)in_sizes; (void)n_in; (void)out_size; (void)ws_size;
  const float* x = (const float*)d_in[0];
  float* out = (float*)d_out;

  // Workspace: t1 (reused as tA) + tB, each N*H*W f2 = 128 MB.
  const size_t planeFloats = (size_t)N_ * H_ * W_ * 2;
  float* t1 = (float*)d_ws;
  float* tB = t1 + planeFloats;

  hbox1_kernel<<<N_ * H_, 64, 0, stream>>>(x, t1);
  vbox1_kernel<<<dim3(W_ / 256, H_ / 256, N_), 256, 0, stream>>>(t1, x, out);
  hbox2_kernel<<<N_ * H_, 64, 0, stream>>>(out, t1, tB);   // t1 reused as tA
  vbox2_kernel<<<dim3(W_ / 256, H_ / 256, N_), 256, 0, stream>>>(t1, tB, out);
}